// LinearAttentionLayer_28106265985635
// MI455X (gfx1250) — compile-verified
//
#include <hip/hip_runtime.h>
#include <hip/hip_bf16.h>
#include <math.h>

typedef float v2f __attribute__((ext_vector_type(2)));
typedef float v8f __attribute__((ext_vector_type(8)));

#define DIM 64
#define SLOPE 0.2f
#define EPS 1e-6f
#define ROWS_PER_BLOCK 64   // 4 waves x 16 rows

// ---------------------------------------------------------------------------
// Kernel 1: per-node LN1 + three GEMMs (WMMA f32 16x16x4) + s1/s2 scores.
//   dense = LN(input)@Ww + Wb          -> ws
//   s1    = tanh(rowsum((q@W1w+W1b)*q)) -> ws
//   s2    = tanh(rowsum((x@W2w+W2b)*x)) -> ws
// One wave computes a 16-row tile; block = 4 waves = 64 rows.
// ---------------------------------------------------------------------------
__global__ __launch_bounds__(128) void node_kernel(
    const float* __restrict__ query, const float* __restrict__ input_,
    const float* __restrict__ Ww,  const float* __restrict__ Wb,
    const float* __restrict__ W1w, const float* __restrict__ W1b,
    const float* __restrict__ W2w, const float* __restrict__ W2b,
    const float* __restrict__ g1,  const float* __restrict__ b1,
    float* __restrict__ dense, float* __restrict__ s1, float* __restrict__ s2,
    int N)
{
    __shared__ float sW[3][DIM][DIM + 1];          // ~50 KB (padded vs bank conflicts)
    __shared__ float sX[ROWS_PER_BLOCK][DIM + 1];  // LN'd input rows
    __shared__ float sQ[ROWS_PER_BLOCK][DIM + 1];  // raw query rows

    // cooperative weight load (weights are L2-resident across blocks)
    {
        const float* Ws[3] = {Ww, W1w, W2w};
        for (int m = 0; m < 3; ++m)
            for (int idx = threadIdx.x; idx < DIM * DIM; idx += blockDim.x)
                sW[m][idx >> 6][idx & 63] = Ws[m][idx];
    }

    const int wave = threadIdx.x >> 5;
    const int lane = threadIdx.x & 31;
    const int half = lane >> 4;          // 0 / 1
    const int l16  = lane & 15;
    const int lrow = wave * 16 + l16;    // local row 0..63
    const int grow = blockIdx.x * ROWS_PER_BLOCK + lrow;
    const int srcrow = grow < N ? grow : (N - 1);

    // ---- LayerNorm(input) : each lane owns half a row (32 floats) ----
    {
        const float4* irow = (const float4*)(input_ + (size_t)srcrow * DIM) + half * 8;
        const float4* qrow = (const float4*)(query  + (size_t)srcrow * DIM) + half * 8;
        float4 iv[8], qv[8];
        float sum = 0.f, sq = 0.f;
        #pragma unroll
        for (int j = 0; j < 8; ++j) {
            iv[j] = irow[j]; qv[j] = qrow[j];
            sum += iv[j].x + iv[j].y + iv[j].z + iv[j].w;
            sq  += iv[j].x*iv[j].x + iv[j].y*iv[j].y + iv[j].z*iv[j].z + iv[j].w*iv[j].w;
        }
        sum += __shfl_xor(sum, 16, 32);   // combine the two half-rows (wave32)
        sq  += __shfl_xor(sq, 16, 32);
        const float mu  = sum * (1.f / DIM);
        const float var = sq * (1.f / DIM) - mu * mu;
        const float rs  = rsqrtf(var + EPS);
        #pragma unroll
        for (int j = 0; j < 8; ++j) {
            const int c = half * 32 + j * 4;
            const float4 gg = *(const float4*)(g1 + c);
            const float4 bb = *(const float4*)(b1 + c);
            sX[lrow][c + 0] = (iv[j].x - mu) * rs * gg.x + bb.x;
            sX[lrow][c + 1] = (iv[j].y - mu) * rs * gg.y + bb.y;
            sX[lrow][c + 2] = (iv[j].z - mu) * rs * gg.z + bb.z;
            sX[lrow][c + 3] = (iv[j].w - mu) * rs * gg.w + bb.w;
            sQ[lrow][c + 0] = qv[j].x;
            sQ[lrow][c + 1] = qv[j].y;
            sQ[lrow][c + 2] = qv[j].z;
            sQ[lrow][c + 3] = qv[j].w;
        }
    }
    __syncthreads();

    // ---- WMMA: 16x16 tiles, K=64 in steps of 4 (V_WMMA_F32_16X16X4_F32) ----
    const int arow = wave * 16 + l16;    // A-matrix row held by this lane
    float part1[8], part2[8];
    #pragma unroll
    for (int r = 0; r < 8; ++r) { part1[r] = 0.f; part2[r] = 0.f; }

    for (int nt = 0; nt < 4; ++nt) {
        const int col = nt * 16 + l16;   // output column held by this lane
        v8f accD = {}, acc1 = {}, acc2 = {};
        #pragma unroll
        for (int k = 0; k < DIM; k += 4) {
            const int ka = k + 2 * half; // A lanes 16-31 carry K=2,3
            v2f ax, aq, bD, bA, bB;
            ax[0] = sX[arow][ka];     ax[1] = sX[arow][ka + 1];
            aq[0] = sQ[arow][ka];     aq[1] = sQ[arow][ka + 1];
            bD[0] = sW[0][ka][col];   bD[1] = sW[0][ka + 1][col];
            bA[0] = sW[1][ka][col];   bA[1] = sW[1][ka + 1][col];
            bB[0] = sW[2][ka][col];   bB[1] = sW[2][ka + 1][col];
            accD = __builtin_amdgcn_wmma_f32_16x16x4_f32(false, ax, false, bD, (short)0, accD, false, false);
            acc1 = __builtin_amdgcn_wmma_f32_16x16x4_f32(false, aq, false, bA, (short)0, acc1, false, false);
            acc2 = __builtin_amdgcn_wmma_f32_16x16x4_f32(false, ax, false, bB, (short)0, acc2, false, false);
        }
        const float biasD = Wb[col], bias1 = W1b[col], bias2 = W2b[col];
        #pragma unroll
        for (int r = 0; r < 8; ++r) {
            const int mloc = wave * 16 + r + 8 * half;   // local row of acc elem r
            const int gr   = blockIdx.x * ROWS_PER_BLOCK + mloc;
            const float dv = accD[r] + biasD;
            if (gr < N) dense[(size_t)gr * DIM + col] = dv;
            part1[r] += (acc1[r] + bias1) * sQ[mloc][col];
            part2[r] += (acc2[r] + bias2) * sX[mloc][col];
        }
    }

    // reduce score partials across the 16 lanes of each half-wave
    #pragma unroll
    for (int r = 0; r < 8; ++r) {
        float p1 = part1[r], p2 = part2[r];
        #pragma unroll
        for (int m = 1; m < 16; m <<= 1) {
            p1 += __shfl_xor(p1, m, 32);
            p2 += __shfl_xor(p2, m, 32);
        }
        const int gr = blockIdx.x * ROWS_PER_BLOCK + wave * 16 + r + 8 * half;
        if (l16 == 0 && gr < N) {
            s1[gr] = tanhf(p1);
            s2[gr] = tanhf(p2);
        }
    }
}

// ---------------------------------------------------------------------------
// Kernel 2: CSR row pointers via binary search on sorted edge_row.
// ---------------------------------------------------------------------------
__global__ __launch_bounds__(256) void rowptr_kernel(
    const int* __restrict__ edge_row, int* __restrict__ rowptr, int N, int E)
{
    const int i = blockIdx.x * blockDim.x + threadIdx.x;
    if (i > N) return;
    int lo = 0, hi = E;
    while (lo < hi) {
        const int mid = (lo + hi) >> 1;
        lo = (edge_row[mid] < i) ? (mid + 1) : lo;
        hi = (edge_row[mid] < i) ? hi : mid;
    }
    rowptr[i] = lo;
}

// ---------------------------------------------------------------------------
// Kernel 3: fused per-row softmax + SpMM + LN2. One wave per row; 32 lanes
// each own 2 of the 64 feature columns (float2), so one edge = one step.
// ---------------------------------------------------------------------------
__global__ __launch_bounds__(256) void row_attn_kernel(
    const int* __restrict__ rowptr, const int* __restrict__ edge_col,
    const float* __restrict__ s1, const float* __restrict__ s2,
    const float* __restrict__ dense,
    const float* __restrict__ g2, const float* __restrict__ b2,
    float* __restrict__ out, int N)
{
    const int wave = threadIdx.x >> 5;
    const int lane = threadIdx.x & 31;
    const int row  = blockIdx.x * 8 + wave;
    if (row >= N) return;

    const int beg = rowptr[row];
    const int end = rowptr[row + 1];
    const float s1r = s1[row];

    // pass 1: row max of leaky_relu(s1[i]+s2[j]) (lanes stride the edges)
    float mx = -3.402823466e38f;
    for (int e = beg + lane; e < end; e += 32) {
        float l = s1r + s2[edge_col[e]];
        l = l > 0.f ? l : SLOPE * l;
        mx = fmaxf(mx, l);
    }
    #pragma unroll
    for (int m = 16; m >= 1; m >>= 1) mx = fmaxf(mx, __shfl_xor(mx, m, 32));

    // pass 2: exp-sum + weighted gather of dense[col] (L2-resident)
    float2 acc = make_float2(0.f, 0.f);
    float den = 0.f;
    for (int e = beg; e < end; ++e) {
        const int col = edge_col[e];           // uniform across the wave
        if (e + 4 < end) __builtin_prefetch(&edge_col[e + 4], 0, 0);
        float l = s1r + s2[col];
        l = l > 0.f ? l : SLOPE * l;
        const float ex = __expf(l - mx);
        den += ex;
        const float2 dv = ((const float2*)(dense + (size_t)col * DIM))[lane];
        acc.x += ex * dv.x;
        acc.y += ex * dv.y;
    }
    const float inv = den > 0.f ? 1.f / den : 0.f;
    acc.x *= inv; acc.y *= inv;

    // fused LayerNorm2 over the 64 features held across the wave
    float sum = acc.x + acc.y;
    float sq  = acc.x * acc.x + acc.y * acc.y;
    #pragma unroll
    for (int m = 16; m >= 1; m >>= 1) {
        sum += __shfl_xor(sum, m, 32);
        sq  += __shfl_xor(sq, m, 32);
    }
    const float mu  = sum * (1.f / DIM);
    const float var = sq * (1.f / DIM) - mu * mu;
    const float rs  = rsqrtf(var + EPS);

    const int c = lane * 2;
    const float2 gg = *(const float2*)(g2 + c);
    const float2 bb = *(const float2*)(b2 + c);
    float2 o;
    o.x = (acc.x - mu) * rs * gg.x + bb.x;
    o.y = (acc.y - mu) * rs * gg.y + bb.y;
    *(float2*)(out + (size_t)row * DIM + c) = o;
}

// ---------------------------------------------------------------------------
extern "C" void kernel_launch(void* const* d_in, const int* in_sizes, int n_in,
                              void* d_out, int out_size, void* d_ws, size_t ws_size,
                              hipStream_t stream) {
    const float* query  = (const float*)d_in[0];
    const float* input_ = (const float*)d_in[1];
    const int*   erow   = (const int*)d_in[2];
    const int*   ecol   = (const int*)d_in[3];
    const float* Ww  = (const float*)d_in[4];
    const float* Wb  = (const float*)d_in[5];
    const float* W1w = (const float*)d_in[6];
    const float* W1b = (const float*)d_in[7];
    const float* W2w = (const float*)d_in[8];
    const float* W2b = (const float*)d_in[9];
    const float* g1  = (const float*)d_in[10];
    const float* b1  = (const float*)d_in[11];
    const float* g2  = (const float*)d_in[12];
    const float* b2  = (const float*)d_in[13];
    float* out = (float*)d_out;

    const int N = in_sizes[0] / DIM;
    const int E = in_sizes[2];

    // workspace layout: dense[N*64] | s1[N] | s2[N] | rowptr[N+1]  (~27 MB)
    float* dense = (float*)d_ws;
    float* s1v   = dense + (size_t)N * DIM;
    float* s2v   = s1v + N;
    int*   rowptr = (int*)(s2v + N);

    node_kernel<<<(N + ROWS_PER_BLOCK - 1) / ROWS_PER_BLOCK, 128, 0, stream>>>(
        query, input_, Ww, Wb, W1w, W1b, W2w, W2b, g1, b1,
        dense, s1v, s2v, N);

    rowptr_kernel<<<(N + 1 + 255) / 256, 256, 0, stream>>>(erow, rowptr, N, E);

    row_attn_kernel<<<(N + 7) / 8, 256, 0, stream>>>(
        rowptr, ecol, s1v, s2v, dense, g2, b2, out, N);
}